// FastAttention_46067819217076
// MI455X (gfx1250) — compile-verified
//
#include <hip/hip_runtime.h>

// Performer linear attention for MI455X (gfx1250, wave32).
// B=4 H=16 N=4096 D=E=64. HBM-bound (~270MB moved, ~4.3 GFLOP) -> use
// full-precision V_WMMA_F32_16X16X4_F32, single-pass HBM traffic, LDS staging.

typedef __attribute__((ext_vector_type(2))) float v2f;
typedef __attribute__((ext_vector_type(4))) float v4f;
typedef __attribute__((ext_vector_type(8))) float v8f;

#define BHN   64          // B*H heads
#define NSEQ  4096
#define DDIM  64
#define EDIM  64
#define SPLIT 8           // N-splits for phase-1 parallelism (512 blocks)
#define NPB   (NSEQ / SPLIT)   // 512 rows per phase-1 block
#define NC    64          // rows staged per chunk
#define PAD   80          // LDS row stride (floats): 2*80 % 64banks == 32 -> no conflicts
#define QPAD  145         // transposed-Q LDS stride: 4*145%64==4 (store), 2*145%64==34 (read)
#define HEADSZ (DDIM * EDIM + DDIM)  // 4096 context + 64 ksum
#define NT    128         // rows per phase-2 block

// ---------------------------------------------------------------------------
// Phase 1: partial context[d][e] = sum_n K[n][d]*V[n][e], partial ksum[d]
// ---------------------------------------------------------------------------
__global__ __launch_bounds__(256) void k1_partial(const float* __restrict__ k,
                                                  const float* __restrict__ v,
                                                  float* __restrict__ part) {
    __shared__ float kl[NC][PAD];
    __shared__ float vl[NC][PAD];
    __shared__ float red[4][DDIM];

    const int blk  = blockIdx.x;          // 0 .. BHN*SPLIT-1
    const int bh   = blk >> 3;
    const int s    = blk & (SPLIT - 1);
    const int tid  = threadIdx.x;
    const int wave = tid >> 5;
    const int lane = tid & 31;
    const int half = lane >> 4;           // K-half of the fragment
    const int l16  = lane & 15;

    const float* kp = k + (size_t)bh * NSEQ * DDIM + (size_t)s * NPB * DDIM;
    const float* vp = v + (size_t)bh * NSEQ * DDIM + (size_t)s * NPB * DDIM;

    // 16 tiles (4 d-blocks x 4 e-blocks) over 8 waves -> 2 tiles/wave
    const int dbase = (wave & 3) * 16;
    const int e0    = (wave >> 2) * 32;

    v8f acc0 = {};
    v8f acc1 = {};
    float ks_acc = 0.f;
    const int sd = tid & 63;              // d index this thread reduces for ksum
    const int sg = tid >> 6;              // row group (0..3)

    const int srow = tid >> 4;            // staging row step
    const int scol = (tid & 15) * 4;      // staging col (float4)

    for (int c = 0; c < NPB; c += NC) {
        // ---- stage K,V chunk (coalesced b128 loads) ----
        #pragma unroll
        for (int r = 0; r < NC; r += 16) {
            const size_t g = (size_t)(c + r + srow) * DDIM + scol;
            *(v4f*)&kl[r + srow][scol] = *(const v4f*)(kp + g);
            *(v4f*)&vl[r + srow][scol] = *(const v4f*)(vp + g);
        }
        if (c + NC < NPB) {               // global_prefetch_b8 next chunk
            __builtin_prefetch(kp + (size_t)(c + NC) * DDIM + tid * 16, 0, 0);
            __builtin_prefetch(vp + (size_t)(c + NC) * DDIM + tid * 16, 0, 0);
        }
        __syncthreads();

        // ---- ksum partial (conflict-free: lanes hit distinct banks) ----
        #pragma unroll
        for (int r = 0; r < NC; r += 4) ks_acc += kl[r + sg][sd];

        // ---- WMMA: 16 K-steps of 4 over this chunk ----
        #pragma unroll 4
        for (int ks = 0; ks < NC; ks += 4) {
            const int r0 = ks + half * 2;
            v2f a, b0, b1;
            a.x  = kl[r0][dbase + l16];
            a.y  = kl[r0 + 1][dbase + l16];
            b0.x = vl[r0][e0 + l16];
            b0.y = vl[r0 + 1][e0 + l16];
            b1.x = vl[r0][e0 + 16 + l16];
            b1.y = vl[r0 + 1][e0 + 16 + l16];
            acc0 = __builtin_amdgcn_wmma_f32_16x16x4_f32(false, a, false, b0,
                                                         (short)0, acc0, false, false);
            acc1 = __builtin_amdgcn_wmma_f32_16x16x4_f32(false, a, false, b1,
                                                         (short)0, acc1, false, false);
        }
        __syncthreads();
    }

    // ---- write partial ksum ----
    red[sg][sd] = ks_acc;
    __syncthreads();
    float* pc = part + ((size_t)s * BHN + bh) * HEADSZ;
    if (tid < DDIM)
        pc[DDIM * EDIM + tid] = red[0][tid] + red[1][tid] + red[2][tid] + red[3][tid];

    // ---- write partial context tiles (C layout: VGPR i -> m=i(+8), lane -> n) ----
    #pragma unroll
    for (int i = 0; i < 8; ++i) {
        const int m = dbase + i + half * 8;
        pc[m * EDIM + e0 + l16]      = acc0[i];
        pc[m * EDIM + e0 + 16 + l16] = acc1[i];
    }
}

// ---------------------------------------------------------------------------
// Phase 1b: reduce SPLIT partials -> final context+ksum per head
// ---------------------------------------------------------------------------
__global__ __launch_bounds__(256) void k1_reduce(const float* __restrict__ part,
                                                 float* __restrict__ fin) {
    const int idx = blockIdx.x * 256 + threadIdx.x;
    if (idx >= BHN * HEADSZ) return;
    const int bh  = idx / HEADSZ;
    const int off = idx - bh * HEADSZ;
    float sacc = 0.f;
    #pragma unroll
    for (int p = 0; p < SPLIT; ++p)
        sacc += part[((size_t)p * BHN + bh) * HEADSZ + off];
    fin[idx] = sacc;
}

// ---------------------------------------------------------------------------
// Phase 2: out[n][e] = (sum_d Q[n][d]*context[d][e]) * Dinv[n]
// ---------------------------------------------------------------------------
__global__ __launch_bounds__(256) void k2_out(const float* __restrict__ q,
                                              const float* __restrict__ fin,
                                              float* __restrict__ out) {
    __shared__ float qlT[DDIM][QPAD];   // Q tile transposed: [d][row], 37.1KB
    __shared__ float cl[DDIM][PAD];     // context [d][e], 20KB
    __shared__ float ksl[DDIM];
    __shared__ float dinv[NT];

    const int blk  = blockIdx.x;        // 0 .. BHN*(NSEQ/NT)-1
    const int bh   = blk >> 5;          // NSEQ/NT == 32
    const int n0   = (blk & 31) * NT;
    const int tid  = threadIdx.x;
    const int wave = tid >> 5;
    const int lane = tid & 31;
    const int half = lane >> 4;
    const int l16  = lane & 15;

    const float* qp = q + ((size_t)bh * NSEQ + n0) * DDIM;
    const float* cp = fin + (size_t)bh * HEADSZ;

    // ---- stage context (row-major, coalesced) ----
    const int srow = tid >> 4;
    const int scol = (tid & 15) * 4;
    #pragma unroll
    for (int r = 0; r < DDIM; r += 16)
        *(v4f*)&cl[r + srow][scol] = *(const v4f*)(cp + (size_t)(r + srow) * EDIM + scol);
    if (tid < DDIM) ksl[tid] = cp[DDIM * EDIM + tid];

    // ---- stage Q transposed (coalesced global b128, conflict-free scatter) ----
    #pragma unroll
    for (int r = 0; r < NT; r += 16) {
        const v4f qv = *(const v4f*)(qp + (size_t)(r + srow) * DDIM + scol);
        qlT[scol + 0][r + srow] = qv.x;
        qlT[scol + 1][r + srow] = qv.y;
        qlT[scol + 2][r + srow] = qv.z;
        qlT[scol + 3][r + srow] = qv.w;
    }
    __syncthreads();

    // ---- per-row normalizer: Dinv[n] = 1 / (q[n] . ksum) ----
    if (tid < NT) {
        float dot = 0.f;
        #pragma unroll
        for (int d0 = 0; d0 < DDIM; ++d0) dot += qlT[d0][tid] * ksl[d0];
        dinv[tid] = 1.0f / dot;
    }

    // ---- WMMA: wave owns 16 rows x full E (4 tiles), K = D = 64 ----
    const int m0 = wave * 16;
    v8f acc[4] = {};
    #pragma unroll 4
    for (int ks = 0; ks < DDIM; ks += 4) {
        const int r0 = ks + half * 2;
        v2f a;
        a.x = qlT[r0][m0 + l16];
        a.y = qlT[r0 + 1][m0 + l16];
        #pragma unroll
        for (int eb = 0; eb < 4; ++eb) {
            v2f b;
            b.x = cl[r0][eb * 16 + l16];
            b.y = cl[r0 + 1][eb * 16 + l16];
            acc[eb] = __builtin_amdgcn_wmma_f32_16x16x4_f32(false, a, false, b,
                                                            (short)0, acc[eb], false, false);
        }
    }
    __syncthreads();   // dinv[] visible to all waves

    // ---- scale + store ----
    float* op = out + ((size_t)bh * NSEQ + n0) * EDIM;
    #pragma unroll
    for (int i = 0; i < 8; ++i) {
        const int m  = m0 + i + half * 8;
        const float di = dinv[m];
        #pragma unroll
        for (int eb = 0; eb < 4; ++eb)
            op[(size_t)m * EDIM + eb * 16 + l16] = acc[eb][i] * di;
    }
}

// ---------------------------------------------------------------------------
extern "C" void kernel_launch(void* const* d_in, const int* in_sizes, int n_in,
                              void* d_out, int out_size, void* d_ws, size_t ws_size,
                              hipStream_t stream) {
    (void)in_sizes; (void)n_in; (void)out_size; (void)ws_size;
    const float* q = (const float*)d_in[0];
    const float* k = (const float*)d_in[1];
    const float* v = (const float*)d_in[2];
    float* out  = (float*)d_out;
    float* part = (float*)d_ws;                             // SPLIT*BHN*HEADSZ floats
    float* fin  = part + (size_t)SPLIT * BHN * HEADSZ;      // BHN*HEADSZ floats

    k1_partial<<<BHN * SPLIT, 256, 0, stream>>>(k, v, part);
    k1_reduce<<<(BHN * HEADSZ + 255) / 256, 256, 0, stream>>>(part, fin);
    k2_out<<<BHN * (NSEQ / NT), 256, 0, stream>>>(q, fin, out);
}